// CQT_7619271983114
// MI455X (gfx1250) — compile-verified
//
#include <hip/hip_runtime.h>
#include <math.h>

// CQT via V_WMMA_F32_16X16X4_F32 (CDNA5, wave32).
// GEMM: C[frame, bin] = sum_l frames[frame, l] * k{r,i}[bin, l]
//   M = 516 frames (4 bt * 129), N = 252 bins, K = L (~69376).
// Memory-bound on streaming kr/ki (~140 MB, one pass) at 23.3 TB/s -> ~6 us floor.
// Each wave: 2 frame tiles (share B frags) x 1 bin tile x (re+im), K split 4 ways.

typedef float v2f __attribute__((ext_vector_type(2)));
typedef float v8f __attribute__((ext_vector_type(8)));

#define HOP_    512
#define TLEN    65536
#define NBINS   252
#define NFRAMES 129     // 1 + TLEN/HOP
#define KSPLIT  4
#define NFPAIR  5       // ceil(9 frame tiles / 2)
#define NTASKS  (4 * NFPAIR * 16)   // 320

#define WMMA_F32(a, b, c) \
    __builtin_amdgcn_wmma_f32_16x16x4_f32(false, (a), false, (b), (short)0, (c), false, false)

__device__ __forceinline__ v2f load_a_edge(const float* __restrict__ xrow,
                                           int aoff, int k) {
    // 32-bit clamped load + select-zero (v_med3 + cndmask, no EXEC change)
    const int i0 = aoff + k;
    const int i1 = i0 + 1;
    const int c0 = min(max(i0, 0), TLEN - 1);
    const int c1 = min(max(i1, 0), TLEN - 1);
    float a0 = xrow[c0];
    float a1 = xrow[c1];
    a0 = (i0 == c0) ? a0 : 0.f;
    a1 = (i1 == c1) ? a1 : 0.f;
    v2f a = {a0, a1};
    return a;
}

__device__ __forceinline__ v2f ld2(const float* p) {
    const float2 t = *(const float2*)p;
    v2f v = {t.x, t.y};
    return v;
}

__global__ __launch_bounds__(256) void cqt_wmma_f32(
    const float* __restrict__ x,    // [4][65536]  (b*2+tr major)
    const float* __restrict__ kr,   // [252][L]
    const float* __restrict__ ki,   // [252][L]
    float* __restrict__ out,        // [2][252][129][2]
    int L)
{
    // [wave][mat: re0,im0,re1,im1][vgpr][lane] = 32 KB
    __shared__ float red[8][4][8][32];

    const int tid    = threadIdx.x;
    const int lane   = tid & 31;
    const int wave   = tid >> 5;                      // 0..7
    const int task   = blockIdx.x * 2 + (wave >> 2);  // 2 tile-tasks per block
    const int ksplit = wave & (KSPLIT - 1);

    // task -> (bt, frame-tile pair, bin tile)
    const int bt    = task / (NFPAIR * 16);
    const int rem   = task - bt * (NFPAIR * 16);
    const int fp    = rem >> 4;                 // 0..4
    const int btile = rem & 15;                 // 0..15
    const int b     = bt >> 1;
    const int tr    = bt & 1;

    const int f0 = 2 * fp;                      // 0,2,4,6,8
    const int f1 = min(2 * fp + 1, 8);          // 1,3,5,7,8 (8 duplicated: benign)

    const int row = lane & 15;                  // matrix row/col within lane half
    const int kh  = lane >> 4;                  // K-pair owned by this lane half
    const int Lh  = L >> 1;

    const int n0_0  = f0 * 16;
    const int n0_1  = f1 * 16;
    const int nrow0 = min(n0_0 + row, NFRAMES - 1);
    const int nrow1 = min(n0_1 + row, NFRAMES - 1);
    const int bin   = min(btile * 16 + row, NBINS - 1);

    const float* xrow = x + (size_t)bt * TLEN;
    // A element (m=row, K=k+kh*2+j) at xrow[nrow*HOP + k - Lh + kh*2 + j]
    const int aoff0 = nrow0 * HOP_ - Lh + kh * 2;
    const int aoff1 = nrow1 * HOP_ - Lh + kh * 2;
    // B element (K, col=bin) at k{r,i}[bin*L + k + kh*2 + j]
    const float* brp = kr + (size_t)bin * L + kh * 2;
    const float* bip = ki + (size_t)bin * L + kh * 2;

    // k range where every A row of BOTH tiles is in-bounds (uniform per wave)
    const int nmax = min(n0_1 + 15, NFRAMES - 1);
    int klo = Lh - n0_0 * HOP_;
    int khi = TLEN + Lh - nmax * HOP_;
    if (klo < 0) klo = 0;
    if (klo > L) klo = L;
    if (khi < klo) khi = klo;
    if (khi > L) khi = L;
    klo = (klo + 3) & ~3;
    khi = khi & ~3;
    if (khi < klo) khi = klo;

    // this wave's k chunk (L multiple of 16 -> L/4 multiple of 4)
    const int kchunk = L >> 2;
    const int kb = ksplit * kchunk;
    const int ke = kb + kchunk;

    v8f cr0 = {0.f,0.f,0.f,0.f,0.f,0.f,0.f,0.f};
    v8f ci0 = {0.f,0.f,0.f,0.f,0.f,0.f,0.f,0.f};
    v8f cr1 = {0.f,0.f,0.f,0.f,0.f,0.f,0.f,0.f};
    v8f ci1 = {0.f,0.f,0.f,0.f,0.f,0.f,0.f,0.f};

    // ---- edge region 1: k in [kb, min(klo,ke)), clamped A (left zero pad) ----
    {
        const int e1 = min(klo, ke);
#pragma unroll 2
        for (int k = kb; k < e1; k += 4) {
            const v2f a0 = load_a_edge(xrow, aoff0, k);
            const v2f a1 = load_a_edge(xrow, aoff1, k);
            const v2f br = ld2(brp + k);
            const v2f bi = ld2(bip + k);
            cr0 = WMMA_F32(a0, br, cr0);
            ci0 = WMMA_F32(a0, bi, ci0);
            cr1 = WMMA_F32(a1, br, cr1);
            ci1 = WMMA_F32(a1, bi, ci1);
        }
    }

    // ---- main region: fixed-trip unrolled chunks, immediate-offset b64 loads ----
    {
        const int mlo = max(klo, kb);
        const int mhi = min(khi, ke);
        int k = mlo;
        const float* ap0 = xrow + ((long)aoff0 + mlo);  // >= xrow in safe region
        const float* ap1 = xrow + ((long)aoff1 + mlo);
        const float* bR  = brp + mlo;
        const float* bI  = bip + mlo;

        for (; k + 32 <= mhi; k += 32) {
            // pull the HBM kernel streams ahead (global_prefetch_b8)
            __builtin_prefetch(bR + 1024, 0, 0);
            __builtin_prefetch(bI + 1024, 0, 0);
#pragma unroll
            for (int u = 0; u < 8; ++u) {
                const v2f a0 = ld2(ap0 + 4 * u);
                const v2f a1 = ld2(ap1 + 4 * u);
                const v2f br = ld2(bR + 4 * u);
                const v2f bi = ld2(bI + 4 * u);
                cr0 = WMMA_F32(a0, br, cr0);
                ci0 = WMMA_F32(a0, bi, ci0);
                cr1 = WMMA_F32(a1, br, cr1);
                ci1 = WMMA_F32(a1, bi, ci1);
            }
            ap0 += 32; ap1 += 32; bR += 32; bI += 32;
        }
        for (; k < mhi; k += 4) {
            const v2f a0 = ld2(ap0);
            const v2f a1 = ld2(ap1);
            const v2f br = ld2(bR);
            const v2f bi = ld2(bI);
            cr0 = WMMA_F32(a0, br, cr0);
            ci0 = WMMA_F32(a0, bi, ci0);
            cr1 = WMMA_F32(a1, br, cr1);
            ci1 = WMMA_F32(a1, bi, ci1);
            ap0 += 4; ap1 += 4; bR += 4; bI += 4;
        }
    }

    // ---- edge region 2: k in [max(khi,kb), ke), clamped A (right zero pad) ----
    {
        const int e3 = max(khi, kb);
#pragma unroll 2
        for (int k = e3; k < ke; k += 4) {
            const v2f a0 = load_a_edge(xrow, aoff0, k);
            const v2f a1 = load_a_edge(xrow, aoff1, k);
            const v2f br = ld2(brp + k);
            const v2f bi = ld2(bip + k);
            cr0 = WMMA_F32(a0, br, cr0);
            ci0 = WMMA_F32(a0, bi, ci0);
            cr1 = WMMA_F32(a1, br, cr1);
            ci1 = WMMA_F32(a1, bi, ci1);
        }
    }

    // ---- k-split reduction through LDS, magnitude, store ----
#pragma unroll
    for (int i = 0; i < 8; ++i) {
        red[wave][0][i][lane] = cr0[i];
        red[wave][1][i][lane] = ci0[i];
        red[wave][2][i][lane] = cr1[i];
        red[wave][3][i][lane] = ci1[i];
    }
    __syncthreads();

    if (ksplit == 0) {
        const int g  = wave;               // 0 or 4: base wave of this tile group
        const int bn = btile * 16 + row;
#pragma unroll
        for (int t = 0; t < 2; ++t) {
            const int fbase = (t == 0) ? n0_0 : n0_1;
#pragma unroll
            for (int i = 0; i < 8; ++i) {
                const float r = red[g+0][2*t  ][i][lane] + red[g+1][2*t  ][i][lane] +
                                red[g+2][2*t  ][i][lane] + red[g+3][2*t  ][i][lane];
                const float m = red[g+0][2*t+1][i][lane] + red[g+1][2*t+1][i][lane] +
                                red[g+2][2*t+1][i][lane] + red[g+3][2*t+1][i][lane];
                // D layout: VGPR i -> row i (lanes 0-15) / i+8 (lanes 16-31);
                //           lane%16 -> bin column
                const int fm = fbase + i + 8 * kh;
                if (fm < NFRAMES && bn < NBINS) {
                    out[(((size_t)b * NBINS + bn) * NFRAMES + fm) * 2 + tr] =
                        sqrtf(r * r + m * m);
                }
            }
        }
    }
}

extern "C" void kernel_launch(void* const* d_in, const int* in_sizes, int n_in,
                              void* d_out, int out_size, void* d_ws, size_t ws_size,
                              hipStream_t stream) {
    const float* x  = (const float*)d_in[0];
    const float* kr = (const float*)d_in[1];
    const float* ki = (const float*)d_in[2];
    float* out = (float*)d_out;

    const int L = in_sizes[1] / NBINS;   // kernel length (multiple of 16)

    // 320 tile-tasks * 4 k-splits = 1280 waves; 8 waves (2 tasks) per block.
    const int blocks = NTASKS / 2;       // 160
    hipLaunchKernelGGL(cqt_wmma_f32, dim3(blocks), dim3(256), 0, stream,
                       x, kr, ki, out, L);
}